// RESCAL_89000312308089
// MI455X (gfx1250) — compile-verified
//
#include <hip/hip_runtime.h>

#define DIM     100
#define NREL    1000
#define NPAIR   16384
#define RSTRIDE 112      // N padded to 7*16
#define NTILES  7        // ceil(100/16)

typedef __attribute__((ext_vector_type(2))) float v2f;
typedef __attribute__((ext_vector_type(8))) float v8f;

// ---- CDNA5 async global->LDS copy (ASYNCcnt-tracked, bypasses VGPRs) ----
__device__ __forceinline__ void async_load_b128(unsigned lds_byte_off, const void* gaddr) {
    asm volatile("global_load_async_to_lds_b128 %0, %1, off"
                 :: "v"(lds_byte_off), "v"(gaddr)
                 : "memory");
}
__device__ __forceinline__ void async_wait0() {
    asm volatile("s_wait_asynccnt 0x0" ::: "memory");
}
__device__ __forceinline__ unsigned lds_off(const void* p) {
    // Flat LDS address: low 32 bits are the wave-relative LDS byte offset.
    return (unsigned)(size_t)p;
}

__global__ void zero_kernel(int* __restrict__ p, int n) {
    int i = blockIdx.x * blockDim.x + threadIdx.x;
    if (i < n) p[i] = 0;
}

__global__ void hist_kernel(const int* __restrict__ r_index, int* __restrict__ counts) {
    int p = blockIdx.x * blockDim.x + threadIdx.x;
    if (p < NPAIR) atomicAdd(&counts[r_index[p]], 1);
}

// Single-block exclusive scan over NREL counts (Hillis-Steele in LDS).
__global__ __launch_bounds__(1024) void scan_kernel(const int* __restrict__ counts,
                                                    int* __restrict__ offsets) {
    __shared__ int s[1024];
    int t = threadIdx.x;
    int c = (t < NREL) ? counts[t] : 0;
    s[t] = c;
    __syncthreads();
    for (int d = 1; d < 1024; d <<= 1) {
        int v = (t >= d) ? s[t - d] : 0;
        __syncthreads();
        s[t] += v;
        __syncthreads();
    }
    if (t < NREL) offsets[t] = s[t] - c;   // exclusive prefix
}

__global__ void scatter_kernel(const int* __restrict__ r_index,
                               const int* __restrict__ offsets,
                               int* __restrict__ cursors,
                               int* __restrict__ pairs) {
    int p = blockIdx.x * blockDim.x + threadIdx.x;
    if (p < NPAIR) {
        int r = r_index[p];
        int pos = offsets[r] + atomicAdd(&cursors[r], 1);
        pairs[pos] = p;
    }
}

// One block per relation. R staged once into LDS (async B128), reused across tiles.
// D[M=16 pairs, N=16 cols] accumulated with V_WMMA_F32_16X16X4_F32 over K=100.
__global__ __launch_bounds__(256) void rescal_kernel(
    const int*   __restrict__ h_index,
    const int*   __restrict__ t_index,
    const float* __restrict__ ent_emb,
    const float* __restrict__ rel_mat,
    const int*   __restrict__ counts,
    const int*   __restrict__ offsets,
    const int*   __restrict__ pairs,
    float*       __restrict__ out) {

    __shared__ float Rs[DIM * RSTRIDE];     // R[i][j], j padded with zeros -> 44.8 KB
    __shared__ float Hs[16 * DIM];          // h rows of current 16-pair tile
    __shared__ float Ts[16 * DIM];          // t rows of current 16-pair tile
    __shared__ float partials[NTILES * 16]; // per-(N-tile, row) partial scores
    __shared__ int   pid[16];

    const int rel = blockIdx.x;
    const int cnt = counts[rel];
    if (cnt == 0) return;
    const int base0 = offsets[rel];

    const int tid  = threadIdx.x;
    const int lane = tid & 31;
    const int wave = tid >> 5;
    const int half = lane >> 4;   // 0: lanes 0-15, 1: lanes 16-31
    const int ln16 = lane & 15;

    // ---- Stage R into LDS exactly once: per-lane async B128 transfers.
    // DIM % 4 == 0 so each 4-dword chunk stays inside one row of R.
    const float* Rg = rel_mat + (size_t)rel * (DIM * DIM);
    for (int t = tid; t < (DIM * DIM) / 4; t += 256) {   // 2500 chunks
        int d0 = 4 * t;
        int i = d0 / DIM, j = d0 - i * DIM;
        async_load_b128(lds_off(&Rs[i * RSTRIDE + j]), Rg + d0);
    }
    // Zero the padded columns j = 100..111.
    for (int t = tid; t < DIM * (RSTRIDE - DIM); t += 256) {  // 1200 dwords
        int i = t / (RSTRIDE - DIM);
        int j = DIM + (t - i * (RSTRIDE - DIM));
        Rs[i * RSTRIDE + j] = 0.0f;
    }

    const int ntiles_m = (cnt + 15) >> 4;
    for (int tile = 0; tile < ntiles_m; ++tile) {
        const int base  = base0 + tile * 16;
        int valid = cnt - tile * 16;
        if (valid > 16) valid = 16;

        __syncthreads();   // prior iteration's consumers are done

        if (tid < 16) pid[tid] = (tid < valid) ? pairs[base + tid] : -1;
        // Stage H/T tiles: 16 rows x 25 B128 chunks each.
        for (int t = tid; t < 16 * (DIM / 4); t += 256) {
            int m = t / (DIM / 4);
            int c = 4 * (t - m * (DIM / 4));
            if (m < valid) {
                int p = pairs[base + m];
                async_load_b128(lds_off(&Hs[m * DIM + c]),
                                ent_emb + (size_t)h_index[p] * DIM + c);
                async_load_b128(lds_off(&Ts[m * DIM + c]),
                                ent_emb + (size_t)t_index[p] * DIM + c);
            } else {
                float4 z = make_float4(0.f, 0.f, 0.f, 0.f);
                *(float4*)&Hs[m * DIM + c] = z;
                *(float4*)&Ts[m * DIM + c] = z;
            }
        }
        async_wait0();     // each wave drains its own async transfers (incl. R on tile 0)
        __syncthreads();

        if (wave < NTILES) {
            const int nt = wave;
            const int j  = nt * 16 + ln16;       // global column of this lane
            v8f acc = {0.f, 0.f, 0.f, 0.f, 0.f, 0.f, 0.f, 0.f};

            // K = 100 -> 25 WMMA steps of K=4.
            for (int kk = 0; kk < DIM; kk += 4) {
                const int k0 = kk + 2 * half;    // ISA layout: vgpr0->K=2*half, vgpr1->K=2*half+1
                v2f a, b;
                a.x = Hs[ln16 * DIM + k0];       // A[m=ln16, k0]
                a.y = Hs[ln16 * DIM + k0 + 1];
                b.x = Rs[k0 * RSTRIDE + j];      // B[k0, j]
                b.y = Rs[(k0 + 1) * RSTRIDE + j];
                acc = __builtin_amdgcn_wmma_f32_16x16x4_f32(
                    /*neg_a=*/false, a, /*neg_b=*/false, b,
                    /*c_mod=*/(short)0, acc, /*reuse_a=*/false, /*reuse_b=*/false);
            }

            // D layout: lane holds D[m = v + 8*half, n = ln16] in acc[v].
            // partial[m] = sum_j D[m,j] * t_m[j]; deterministic butterfly reduction.
            #pragma unroll
            for (int v = 0; v < 8; ++v) {
                const int m = v + 8 * half;
                float x = (j < DIM) ? acc[v] * Ts[m * DIM + j] : 0.0f;
                x += __shfl_xor(x, 1);
                x += __shfl_xor(x, 2);
                x += __shfl_xor(x, 4);
                x += __shfl_xor(x, 8);
                if (ln16 == 0) partials[nt * 16 + m] = x;
            }
        }
        __syncthreads();

        if (tid < 16 && pid[tid] >= 0) {
            float s = 0.0f;
            #pragma unroll
            for (int nt = 0; nt < NTILES; ++nt) s += partials[nt * 16 + tid];
            out[pid[tid]] = -s;
        }
    }
}

extern "C" void kernel_launch(void* const* d_in, const int* in_sizes, int n_in,
                              void* d_out, int out_size, void* d_ws, size_t ws_size,
                              hipStream_t stream) {
    const int*   h_index = (const int*)d_in[0];
    const int*   t_index = (const int*)d_in[1];
    const int*   r_index = (const int*)d_in[2];
    const float* ent_emb = (const float*)d_in[3];
    const float* rel_mat = (const float*)d_in[4];
    float*       out     = (float*)d_out;

    // Workspace layout (ints): counts[1024] | cursors[1024] | offsets[1024] | pairs[16384]
    int* counts  = (int*)d_ws;
    int* cursors = counts + 1024;
    int* offsets = cursors + 1024;
    int* pairs   = offsets + 1024;

    zero_kernel<<<(2048 + 255) / 256, 256, 0, stream>>>(counts, 2048); // counts+cursors
    hist_kernel<<<(NPAIR + 255) / 256, 256, 0, stream>>>(r_index, counts);
    scan_kernel<<<1, 1024, 0, stream>>>(counts, offsets);
    scatter_kernel<<<(NPAIR + 255) / 256, 256, 0, stream>>>(r_index, offsets, cursors, pairs);
    rescal_kernel<<<NREL, 256, 0, stream>>>(h_index, t_index, ent_emb, rel_mat,
                                            counts, offsets, pairs, out);
}